// EGRETLayer_1546188226670
// MI455X (gfx1250) — compile-verified
//
#include <hip/hip_runtime.h>
#include <hip/hip_bf16.h>
#include <math.h>
#include <stdint.h>

// ---------------------------------------------------------------------------
// EGRET layer for MI455X (gfx1250, wave32).
//
// Algebraic refactoring:
//   v_eatt = W_eatt^T @ att3,  c_att = b_att + dot(att3, b_eatt)
//   logit_e = s1[src] + s2[dst] + dot(v_eatt, ex[e]) + c_att
//   h[n]    = segsum(alpha*z_src)[n] + W_edge @ segsum(alpha*ex)[n]
//             + 1{deg(n)>0} * b_edge
//
// GEMMs use exact-f32 WMMA V_WMMA_F32_16X16X4_F32; A-tiles staged into LDS
// with GLOBAL_LOAD_ASYNC_TO_LDS_B128 (ASYNCcnt path, no VGPR round-trip).
// Fragment layout assumptions (ISA 7.12.2):
//   A 16x4 : a[v] @ lane l = A[M = l%16, K = 2*(l/16) + v]
//   B  4x16: b[v] @ lane l = B[K = 2*(l/16) + v, N = l%16]
//   C 16x16: c[r] @ lane l = C[M = r + 8*(l/16), N = l%16]
// ---------------------------------------------------------------------------

#define IN_DIM   256
#define OUT_DIM  128
#define EDGE_DIM 64
#define NEG_SLOPE 0.2f

typedef float v2f __attribute__((ext_vector_type(2)));
typedef float v8f __attribute__((ext_vector_type(8)));

__device__ __forceinline__ void atomicMaxFloat(float* addr, float value) {
    if (value >= 0.0f) {
        atomicMax((int*)addr, __float_as_int(value));
    } else {
        atomicMin((unsigned int*)addr, __float_as_uint(value));
    }
}

// 16B global -> LDS async copy (per-lane), tracked by ASYNCcnt.
__device__ __forceinline__ void async_copy16(void* lds_ptr, const void* gptr) {
    uint32_t lds = (uint32_t)(uintptr_t)lds_ptr;   // flat low 32 bits == LDS offset
    uint64_t ga  = (uint64_t)(uintptr_t)gptr;
    asm volatile("global_load_async_to_lds_b128 %0, %1, off"
                 :: "v"(lds), "v"(ga) : "memory");
}
__device__ __forceinline__ void wait_async0() {
    asm volatile("s_wait_asynccnt 0x0" ::: "memory");
}

// ---------------------------------------------------------------------------
// K0: zero / -inf initialize accumulation buffers (h lives in d_out).
// ---------------------------------------------------------------------------
__global__ void k_init(float* __restrict__ h, float* __restrict__ node_max,
                       float* __restrict__ denom, float* __restrict__ agg_ex, int N) {
    const long hN  = (long)N * OUT_DIM;
    const long aN  = (long)N * EDGE_DIM;
    const long tot = hN + 2L * N + aN;
    for (long i = blockIdx.x * (long)blockDim.x + threadIdx.x; i < tot;
         i += (long)gridDim.x * blockDim.x) {
        if (i < hN)               h[i] = 0.0f;
        else if (i < hN + N)      node_max[i - hN] = -3.402823466e+38f;
        else if (i < hN + 2L * N) denom[i - hN - N] = 0.0f;
        else                      agg_ex[i - hN - 2L * N] = 0.0f;
    }
}

// ---------------------------------------------------------------------------
// K0b: fold W_eatt/b_eatt into the attention vector (1 block, 64 threads).
// ---------------------------------------------------------------------------
__global__ void k_prep_att(const float* __restrict__ Watt, const float* __restrict__ batt,
                           const float* __restrict__ Weatt, const float* __restrict__ beatt,
                           float* __restrict__ v_eatt, float* __restrict__ c_att) {
    const float* att3 = Watt + 2 * OUT_DIM;
    const int t = threadIdx.x;  // 0..63
    float s = 0.0f;
    for (int j = 0; j < EDGE_DIM; ++j) s += att3[j] * Weatt[j * EDGE_DIM + t];
    v_eatt[t] = s;
    if (t == 0) {
        float c = batt[0];
        for (int j = 0; j < EDGE_DIM; ++j) c += att3[j] * beatt[j];
        *c_att = c;
    }
}

// ---------------------------------------------------------------------------
// K1: z[N,128] = nfeat[N,256] @ W_fc^T + b_fc     (f32 WMMA, 16x16 tiles)
// Block = 256 threads = 8 waves; wave w owns output cols [16w, 16w+16).
// A tile (16 x 256, pitch 260 -> conflict-free, 16B-aligned rows) staged via
// async-to-LDS B128; tail tiles fall back to a guarded cooperative load.
// ---------------------------------------------------------------------------
__global__ __launch_bounds__(256) void k_node_gemm(const float* __restrict__ nfeat,
                                                   const float* __restrict__ Wfc,
                                                   const float* __restrict__ bfc,
                                                   float* __restrict__ z, int N) {
    __shared__ float As[16][IN_DIM + 4];
    const int m0  = blockIdx.x * 16;
    const int tid = threadIdx.x;

    if (m0 + 16 <= N) {
        // 16*256 floats = 1024 x 16B chunks; 4 per thread, coalesced rows.
#pragma unroll
        for (int i = 0; i < 4; ++i) {
            int chunk = tid + i * 256;       // 0..1023
            int r  = chunk >> 6;             // 64 chunks per row
            int c4 = (chunk & 63) << 2;      // starting float col
            async_copy16(&As[r][c4], nfeat + (size_t)(m0 + r) * IN_DIM + c4);
        }
        wait_async0();
    } else {
        for (int r = 0; r < 16; ++r) {
            int row = m0 + r;
            As[r][tid] = (row < N) ? nfeat[(size_t)row * IN_DIM + tid] : 0.0f;
        }
    }
    __syncthreads();

    const int wave = tid >> 5;
    const int lane = tid & 31;
    const int hl   = lane >> 4;   // half-wave select
    const int lo   = lane & 15;
    const int o0   = wave * 16;

    v8f acc = {};
    const float* wrow = Wfc + (size_t)(o0 + lo) * IN_DIM;
    for (int k = 0; k < IN_DIM; k += 4) {
        v2f a, b;
        a.x = As[lo][k + 2 * hl + 0];
        a.y = As[lo][k + 2 * hl + 1];
        b.x = wrow[k + 2 * hl + 0];
        b.y = wrow[k + 2 * hl + 1];
        acc = __builtin_amdgcn_wmma_f32_16x16x4_f32(false, a, false, b,
                                                    (short)0, acc, false, false);
    }

    const float bias = bfc[o0 + lo];
#pragma unroll
    for (int r = 0; r < 8; ++r) {
        int row = m0 + r + 8 * hl;
        if (row < N) z[(size_t)row * OUT_DIM + o0 + lo] = acc[r] + bias;
    }
}

// ---------------------------------------------------------------------------
// K2: per-node attention contributions s1 = z . att1, s2 = z . att2
// One wave32 per node, lane-strided dot + shuffle reduction.
// ---------------------------------------------------------------------------
__global__ __launch_bounds__(256) void k_node_scores(const float* __restrict__ z,
                                                     const float* __restrict__ Watt,
                                                     float* __restrict__ s1,
                                                     float* __restrict__ s2, int N) {
    const int node = (int)((blockIdx.x * (long)blockDim.x + threadIdx.x) >> 5);
    const int lane = threadIdx.x & 31;
    if (node >= N) return;
    const float* zr = z + (size_t)node * OUT_DIM;
    float a1 = 0.0f, a2 = 0.0f;
#pragma unroll
    for (int i = lane; i < OUT_DIM; i += 32) {
        float zv = zr[i];
        a1 += zv * Watt[i];
        a2 += zv * Watt[OUT_DIM + i];
    }
#pragma unroll
    for (int off = 16; off; off >>= 1) {
        a1 += __shfl_xor(a1, off, 32);
        a2 += __shfl_xor(a2, off, 32);
    }
    if (lane == 0) { s1[node] = a1; s2[node] = a2; }
}

// ---------------------------------------------------------------------------
// K3: per-edge leaky-relu logit + segment-max into node_max. Wave per edge.
// ---------------------------------------------------------------------------
__global__ __launch_bounds__(256) void k_edge_scores(const float* __restrict__ ex,
                                                     const int* __restrict__ src,
                                                     const int* __restrict__ dst,
                                                     const float* __restrict__ s1,
                                                     const float* __restrict__ s2,
                                                     const float* __restrict__ v_eatt,
                                                     const float* __restrict__ c_att,
                                                     float* __restrict__ e_score,
                                                     float* __restrict__ node_max, int E) {
    const int e    = (int)((blockIdx.x * (long)blockDim.x + threadIdx.x) >> 5);
    const int lane = threadIdx.x & 31;
    if (e >= E) return;
    const float2 xv = ((const float2*)(ex + (size_t)e * EDGE_DIM))[lane];
    const float2 vv = ((const float2*)v_eatt)[lane];
    float d = xv.x * vv.x + xv.y * vv.y;
#pragma unroll
    for (int off = 16; off; off >>= 1) d += __shfl_xor(d, off, 32);
    if (lane == 0) {
        float logit = s1[src[e]] + s2[dst[e]] + d + c_att[0];
        float esc = (logit > 0.0f) ? logit : NEG_SLOPE * logit;
        e_score[e] = esc;
        atomicMaxFloat(&node_max[dst[e]], esc);
    }
}

// ---------------------------------------------------------------------------
// K4: w = exp(e - max[dst]); denominator via atomic add. Thread per edge.
// Stores w into the alpha slot of d_out (normalized in K5).
// ---------------------------------------------------------------------------
__global__ __launch_bounds__(256) void k_edge_w(const float* __restrict__ e_score,
                                                const int* __restrict__ dst,
                                                const float* __restrict__ node_max,
                                                float* __restrict__ denom,
                                                float* __restrict__ alpha, int E) {
    const int e = blockIdx.x * blockDim.x + threadIdx.x;
    if (e >= E) return;
    const int d = dst[e];
    const float w = __expf(e_score[e] - node_max[d]);
    alpha[e] = w;
    atomicAdd(&denom[d], w);
}

// ---------------------------------------------------------------------------
// K5: alpha = w/denom[dst]; scatter alpha*z_src into h and alpha*ex into
// agg_ex with native f32 global atomics (h + agg_ex ~38 MB: L2-resident).
// Wave per edge: coalesced 128-wide / 64-wide atomic bursts.
// ---------------------------------------------------------------------------
__global__ __launch_bounds__(256) void k_edge_aggregate(const float* __restrict__ ex,
                                                        const int* __restrict__ src,
                                                        const int* __restrict__ dst,
                                                        const float* __restrict__ z,
                                                        const float* __restrict__ denom,
                                                        float* __restrict__ alpha,
                                                        float* __restrict__ h,
                                                        float* __restrict__ agg_ex, int E) {
    const int e    = (int)((blockIdx.x * (long)blockDim.x + threadIdx.x) >> 5);
    const int lane = threadIdx.x & 31;
    if (e >= E) return;
    const int s = src[e], d = dst[e];
    const float a = alpha[e] / denom[d];
    if (lane == 0) alpha[e] = a;
    const float* zs = z + (size_t)s * OUT_DIM;
    float* hd = h + (size_t)d * OUT_DIM;
#pragma unroll
    for (int i = 0; i < OUT_DIM / 32; ++i)
        atomicAdd(&hd[lane + 32 * i], a * zs[lane + 32 * i]);
    const float* exr = ex + (size_t)e * EDGE_DIM;
    float* ag = agg_ex + (size_t)d * EDGE_DIM;
#pragma unroll
    for (int i = 0; i < EDGE_DIM / 32; ++i)
        atomicAdd(&ag[lane + 32 * i], a * exr[lane + 32 * i]);
}

// ---------------------------------------------------------------------------
// K6: h += agg_ex @ W_edge^T + 1{deg>0} * b_edge   (f32 WMMA, K=64)
// A tile 16 x 64, pitch 68 (16B-aligned rows), async-to-LDS staged.
// ---------------------------------------------------------------------------
__global__ __launch_bounds__(256) void k_node_finalize(const float* __restrict__ agg_ex,
                                                       const float* __restrict__ Wedge,
                                                       const float* __restrict__ bedge,
                                                       const float* __restrict__ denom,
                                                       float* __restrict__ h, int N) {
    __shared__ float As[16][EDGE_DIM + 4];
    const int m0  = blockIdx.x * 16;
    const int tid = threadIdx.x;

    if (m0 + 16 <= N) {
        // 16*64 floats = 256 x 16B chunks; one per thread.
        int r  = tid >> 4;                  // 16 chunks per row
        int c4 = (tid & 15) << 2;
        async_copy16(&As[r][c4], agg_ex + (size_t)(m0 + r) * EDGE_DIM + c4);
        wait_async0();
    } else {
        for (int i = tid; i < 16 * EDGE_DIM; i += 256) {
            int r = i >> 6, c = i & 63;
            int row = m0 + r;
            As[r][c] = (row < N) ? agg_ex[(size_t)row * EDGE_DIM + c] : 0.0f;
        }
    }
    __syncthreads();

    const int wave = tid >> 5;
    const int lane = tid & 31;
    const int hl   = lane >> 4;
    const int lo   = lane & 15;
    const int o0   = wave * 16;

    v8f acc = {};
    const float* wrow = Wedge + (size_t)(o0 + lo) * EDGE_DIM;
    for (int k = 0; k < EDGE_DIM; k += 4) {
        v2f a, b;
        a.x = As[lo][k + 2 * hl + 0];
        a.y = As[lo][k + 2 * hl + 1];
        b.x = wrow[k + 2 * hl + 0];
        b.y = wrow[k + 2 * hl + 1];
        acc = __builtin_amdgcn_wmma_f32_16x16x4_f32(false, a, false, b,
                                                    (short)0, acc, false, false);
    }

    const float bias = bedge[o0 + lo];
#pragma unroll
    for (int r = 0; r < 8; ++r) {
        int row = m0 + r + 8 * hl;
        if (row < N) {
            float ind = (denom[row] > 0.0f) ? 1.0f : 0.0f;
            size_t idx = (size_t)row * OUT_DIM + o0 + lo;
            h[idx] = h[idx] + acc[r] + ind * bias;
        }
    }
}

// ---------------------------------------------------------------------------
// Host-side launch
// ---------------------------------------------------------------------------
extern "C" void kernel_launch(void* const* d_in, const int* in_sizes, int n_in,
                              void* d_out, int out_size, void* d_ws, size_t ws_size,
                              hipStream_t stream) {
    const float* nfeat = (const float*)d_in[0];
    const float* ex    = (const float*)d_in[1];
    const int*   src   = (const int*)d_in[2];
    const int*   dst   = (const int*)d_in[3];
    const float* Wfc   = (const float*)d_in[4];
    const float* bfc   = (const float*)d_in[5];
    const float* Watt  = (const float*)d_in[6];
    const float* batt  = (const float*)d_in[7];
    const float* Wedge = (const float*)d_in[8];
    const float* bedge = (const float*)d_in[9];
    const float* Weatt = (const float*)d_in[10];
    const float* beatt = (const float*)d_in[11];

    const int N = in_sizes[0] / IN_DIM;
    const int E = in_sizes[2];

    float* h     = (float*)d_out;                        // [N, OUT_DIM]
    float* alpha = (float*)d_out + (size_t)N * OUT_DIM;  // [E]

    // workspace carve-up (all f32), each region 16-float (64B) aligned so the
    // async B128 staging of agg_ex stays 16B-aligned for any N/E.
    auto align16 = [](size_t v) { return (v + 15) & ~(size_t)15; };
    float* ws       = (float*)d_ws;
    float* z        = ws;                                           // N*128
    float* s1       = z + align16((size_t)N * OUT_DIM);             // N
    float* s2       = s1 + align16((size_t)N);                      // N
    float* node_max = s2 + align16((size_t)N);                      // N
    float* denom    = node_max + align16((size_t)N);                // N
    float* e_score  = denom + align16((size_t)N);                   // E
    float* agg_ex   = e_score + align16((size_t)E);                 // N*64
    float* v_eatt   = agg_ex + align16((size_t)N * EDGE_DIM);       // 64
    float* c_att    = v_eatt + EDGE_DIM;                            // 1
    (void)ws_size; (void)n_in; (void)out_size;

    const int node_tiles = (N + 15) / 16;
    const int node_waves = (N + 7) / 8;   // blocks of 8 waves, wave per node
    const int edge_waves = (E + 7) / 8;   // blocks of 8 waves, wave per edge
    const int edge_thr   = (E + 255) / 256;

    k_init<<<2048, 256, 0, stream>>>(h, node_max, denom, agg_ex, N);
    k_prep_att<<<1, EDGE_DIM, 0, stream>>>(Watt, batt, Weatt, beatt, v_eatt, c_att);
    k_node_gemm<<<node_tiles, 256, 0, stream>>>(nfeat, Wfc, bfc, z, N);
    k_node_scores<<<node_waves, 256, 0, stream>>>(z, Watt, s1, s2, N);
    k_edge_scores<<<edge_waves, 256, 0, stream>>>(ex, src, dst, s1, s2, v_eatt, c_att,
                                                  e_score, node_max, E);
    k_edge_w<<<edge_thr, 256, 0, stream>>>(e_score, dst, node_max, denom, alpha, E);
    k_edge_aggregate<<<edge_waves, 256, 0, stream>>>(ex, src, dst, z, denom, alpha,
                                                     h, agg_ex, E);
    k_node_finalize<<<node_tiles, 256, 0, stream>>>(agg_ex, Wedge, bedge, denom, h, N);
}